// DFPS_Sampler_5892695130399
// MI455X (gfx1250) — compile-verified
//
#include <hip/hip_runtime.h>
#include <stdint.h>

// ---- problem constants (reference instance: B=8, N=65536, npoint=1024) ----
#define B_BATCH   8
#define N_PTS     65536
#define BPB       4                       // blocks cooperating per batch
#define NTHREADS  512                     // 16 wave32 waves per block
#define PPT       (N_PTS / (BPB * NTHREADS))   // 32 points per thread (register-resident)
#define PTS_PER_BLOCK (NTHREADS * PPT)    // 16384
#define TILE_PTS  2048                    // staged per async tile (24 KB LDS)
#define NTILES    (PTS_PER_BLOCK / TILE_PTS)   // 8
#define PPT_TILE  (TILE_PTS / NTHREADS)   // 4
#define CHUNKS_PER_TILE ((TILE_PTS * 12) / 16)  // 1536 x 16B
#define CHUNKS_PER_THREAD (CHUNKS_PER_TILE / NTHREADS) // 3

__device__ __forceinline__ void argmax2(float& v, int& i, float ov, int oi) {
  // strict > with smaller-index tie-break == jnp.argmax first-occurrence
  if (ov > v || (ov == v && oi < i)) { v = ov; i = oi; }
}

__global__ void fps_init_kernel(unsigned* cnt, unsigned long long* cand) {
  int t = threadIdx.x;
  if (t < B_BATCH) cnt[t] = 0u;
  if (t < B_BATCH * 2 * BPB) cand[t] = 0ull;
}

__global__ __launch_bounds__(NTHREADS)
void fps_main_kernel(const float* __restrict__ points,
                     const int* __restrict__ npoint_p,
                     int* __restrict__ out,
                     unsigned* __restrict__ cnt,
                     unsigned long long* __restrict__ cand) {
  __shared__ float4 sbuf[CHUNKS_PER_TILE];   // 24 KB async staging tile
  __shared__ float  swv[NTHREADS / 32];
  __shared__ int    swi[NTHREADS / 32];
  __shared__ float  qsh[3];

  const int tid  = threadIdx.x;
  const int lane = tid & 31;
  const int wid  = tid >> 5;
  const int blk  = blockIdx.x;   // 0..BPB-1
  const int b    = blockIdx.y;   // batch
  const int npoint = *npoint_p;

  const float* pb = points + (size_t)b * N_PTS * 3;
  const unsigned long long gbase =
      (unsigned long long)(uintptr_t)(pb + (size_t)blk * PTS_PER_BLOCK * 3);
  const unsigned lds_base = (unsigned)(uintptr_t)(void*)&sbuf[0];

  float px[PPT], py[PPT], pz[PPT], md[PPT];

  // ---- stage this block's 192KB point slab to VGPRs via async global->LDS ----
#pragma unroll
  for (int t = 0; t < NTILES; ++t) {
#pragma unroll
    for (int j = 0; j < CHUNKS_PER_THREAD; ++j) {
      const int c = tid + j * NTHREADS;
      const unsigned dst = lds_base + (unsigned)c * 16u;
      const unsigned long long src =
          gbase + (unsigned long long)t * (TILE_PTS * 12) + (unsigned long long)c * 16u;
      asm volatile("global_load_async_to_lds_b128 %0, %1, off"
                   :: "v"(dst), "v"(src) : "memory");
    }
    asm volatile("s_wait_asynccnt 0" ::: "memory");
    __syncthreads();
    const float* sf = (const float*)sbuf;
#pragma unroll
    for (int r = 0; r < PPT_TILE; ++r) {
      const int i = t * PPT_TILE + r;          // compile-time after unroll
      const int o = (tid + r * NTHREADS) * 3;  // in-tile xyz offset
      px[i] = sf[o + 0];
      py[i] = sf[o + 1];
      pz[i] = sf[o + 2];
    }
    __syncthreads();  // tile fully consumed before next overwrite
  }

#pragma unroll
  for (int i = 0; i < PPT; ++i) md[i] = 1e10f;

  // first selected index is 0
  float qx = pb[0], qy = pb[1], qz = pb[2];
  if (blk == 0 && tid == 0) out[(size_t)b * npoint] = 0;

  for (int k = 1; k < npoint; ++k) {
    // ---- update mindist vs q, thread-local argmax ----
    float bv = -1.0f;
    int   bi = 0x7fffffff;
#pragma unroll
    for (int i = 0; i < PPT; ++i) {
      const float dx = px[i] - qx, dy = py[i] - qy, dz = pz[i] - qz;
      const float d = dx * dx + dy * dy + dz * dz;
      md[i] = fminf(md[i], d);
      const int gi = blk * PTS_PER_BLOCK + tid + i * NTHREADS;
      argmax2(bv, bi, md[i], gi);
    }
    // ---- wave32 butterfly argmax ----
#pragma unroll
    for (int off = 16; off > 0; off >>= 1) {
      const float ov = __shfl_xor(bv, off, 32);
      const int   oi = __shfl_xor(bi, off, 32);
      argmax2(bv, bi, ov, oi);
    }
    if (lane == 0) { swv[wid] = bv; swi[wid] = bi; }
    __syncthreads();

    if (wid == 0) {
      float v2 = (lane < NTHREADS / 32) ? swv[lane] : -1.0f;
      int   i2 = (lane < NTHREADS / 32) ? swi[lane] : 0x7fffffff;
#pragma unroll
      for (int off = 16; off > 0; off >>= 1) {
        const float ov = __shfl_xor(v2, off, 32);
        const int   oi = __shfl_xor(i2, off, 32);
        argmax2(v2, i2, ov, oi);
      }
      if (lane == 0) {
        // ---- inter-block combine: release candidate, arrive, spin, reduce ----
        const int par = k & 1;
        unsigned long long* cslot = cand + ((size_t)b * 2 + par) * BPB;
        const unsigned long long packed =
            ((unsigned long long)__float_as_uint(v2) << 32) | (unsigned)i2;
        __hip_atomic_store(&cslot[blk], packed, __ATOMIC_RELAXED,
                           __HIP_MEMORY_SCOPE_AGENT);
        __hip_atomic_fetch_add(&cnt[b], 1u, __ATOMIC_RELEASE,
                               __HIP_MEMORY_SCOPE_AGENT);
        const unsigned target = (unsigned)(BPB * k);
        while (__hip_atomic_load(&cnt[b], __ATOMIC_ACQUIRE,
                                 __HIP_MEMORY_SCOPE_AGENT) < target)
          __builtin_amdgcn_s_sleep(1);

        float wv = -1.0f;
        int   wi = 0x7fffffff;
#pragma unroll
        for (int j = 0; j < BPB; ++j) {
          const unsigned long long p = __hip_atomic_load(
              &cslot[j], __ATOMIC_RELAXED, __HIP_MEMORY_SCOPE_AGENT);
          argmax2(wv, wi, __uint_as_float((unsigned)(p >> 32)),
                  (int)(unsigned)(p & 0xffffffffu));
        }
        const float* wp = pb + (size_t)wi * 3;   // L2-hot gather of winner xyz
        qsh[0] = wp[0]; qsh[1] = wp[1]; qsh[2] = wp[2];
        if (blk == 0) out[(size_t)b * npoint + k] = wi;
      }
    }
    __syncthreads();
    qx = qsh[0]; qy = qsh[1]; qz = qsh[2];
  }
}

extern "C" void kernel_launch(void* const* d_in, const int* in_sizes, int n_in,
                              void* d_out, int out_size, void* d_ws, size_t ws_size,
                              hipStream_t stream) {
  const float* points   = (const float*)d_in[0];
  // d_in[1] = features: unused by D-FPS (matches reference)
  const int*   npoint_p = (const int*)d_in[2];
  int* out = (int*)d_out;

  unsigned* cnt = (unsigned*)d_ws;                                   // B counters
  unsigned long long* cand = (unsigned long long*)((char*)d_ws + 256); // B*2*BPB slots

  fps_init_kernel<<<1, 256, 0, stream>>>(cnt, cand);
  fps_main_kernel<<<dim3(BPB, B_BATCH), NTHREADS, 0, stream>>>(
      points, npoint_p, out, cnt, cand);
}